// PrototypeDiversityLoss_16630113370927
// MI455X (gfx1250) — compile-verified
//
#include <hip/hip_runtime.h>
#include <hip/hip_bf16.h>

typedef __attribute__((ext_vector_type(2))) float v2f;
typedef __attribute__((ext_vector_type(8))) float v8f;

#define NPIX   65536      // H*W
#define BATCH  4
#define DDIM   256
#define NCLS   16
#define NPROT  5
#define NPT    80         // NCLS*NPROT
#define EPSV   1e-10f
#define PROTOB_FLOATS (64*5*64)   // 20480 floats = 80KB, WMMA-B-swizzled protos

// ---- monotonic float<->uint mapping for atomic max over signed floats ----
__device__ __forceinline__ unsigned mapf(float f) {
    unsigned u = __float_as_uint(f);
    return (u & 0x80000000u) ? ~u : (u | 0x80000000u);
}
__device__ __forceinline__ float unmapf(unsigned u) {
    unsigned v = (u & 0x80000000u) ? (u ^ 0x80000000u) : ~u;
    return __uint_as_float(v);
}

// ---- init workspace stats ----
__global__ void k_init(unsigned* gmax, float* gZ1, float* gZ2, int* gcnt,
                       float* gG, float* out) {
    int i = blockIdx.x * blockDim.x + threadIdx.x;
    if (i < BATCH * NPT) { gmax[i] = 0u; gZ1[i] = 0.f; gZ2[i] = 0.f; }
    if (i < BATCH * NCLS) gcnt[i] = 0;
    if (i < BATCH * NCLS * 25) gG[i] = 0.f;
    if (i == 0) out[0] = 0.f;
}

// ---- normalize prototypes + swizzle into WMMA B-matrix lane layout ----
// B tile is 4x16 (KxN): vgpr = k&1, lane = (p%16) + 16*(k>>1)
// stored as float2-per-lane: protoB[(g*5 + t)*64 + lane*2 + vgpr]
__global__ void k_protos(const float* __restrict__ protos, float* __restrict__ protoB) {
    int p = blockIdx.x;      // 0..79
    int d = threadIdx.x;     // 0..255
    float v = protos[p * DDIM + d];
    __shared__ float red[256];
    red[d] = v * v;
    __syncthreads();
    for (int s = 128; s > 0; s >>= 1) {
        if (d < s) red[d] += red[d + s];
        __syncthreads();
    }
    float invn = 1.0f / fmaxf(sqrtf(red[0]), 1e-12f);
    float vn = v * invn;
    int g  = d >> 2, kk = d & 3;
    int vg = kk & 1, half = kk >> 1;
    int t = p >> 4, col = p & 15;
    int lane = col + (half << 4);
    protoB[(g * 5 + t) * 64 + lane * 2 + vg] = vn;
}

// ---- main pass: WMMA f32 16x16x4 dense sims, keep the 5 class columns ----
// One wave handles 32 pixels (two A tiles) x 80 prototypes, K=256 stepped by 4.
// Dual A-tile: each LDS B load feeds two independent WMMAs.
__global__ void k_sims(const float* __restrict__ feat, const float* __restrict__ protoB,
                       const int* __restrict__ gt, float* __restrict__ sims) {
    extern __shared__ float ldsB[];      // 20480 floats (80 KB)
    for (int i = threadIdx.x; i < PROTOB_FLOATS; i += blockDim.x)
        ldsB[i] = protoB[i];
    __syncthreads();

    const int lane   = threadIdx.x & 31;
    const int waveId = blockIdx.x * (blockDim.x >> 5) + (threadIdx.x >> 5);
    const int gpix   = waveId << 5;          // base pixel (linear over B*NPIX), 32/wave
    const int b      = gpix >> 16;
    const int n0     = gpix & (NPIX - 1);
    const int m      = lane & 15;
    const int hi     = lane >> 4;            // 0/1 -> K offset 0/2

    // A-tile element addr: feat[(b*DDIM + d)*NPIX + n], d = 4g + 2*hi + {0,1}
    const float* base0 = feat + ((b * DDIM + 2 * hi) * NPIX + n0 + m);
    const float* base1 = base0 + 16;

    v8f x0 = {}, x1 = {}, x2 = {}, x3 = {}, x4 = {};   // tile0 accumulators
    v8f y0 = {}, y1 = {}, y2 = {}, y3 = {}, y4 = {};   // tile1 accumulators
    float ss0 = 0.f, ss1 = 0.f;

    for (int g = 0; g < 64; ++g) {
        float a0 = base0[(g * 4) * NPIX];
        float a1 = base0[(g * 4 + 1) * NPIX];
        float c0 = base1[(g * 4) * NPIX];
        float c1 = base1[(g * 4 + 1) * NPIX];
        ss0 += a0 * a0 + a1 * a1;
        ss1 += c0 * c0 + c1 * c1;
        v2f a; a.x = a0; a.y = a1;
        v2f c; c.x = c0; c.y = c1;
        const v2f* lb = (const v2f*)&ldsB[g * 320];
        v2f b0 = lb[0 * 32 + lane];
        v2f b1 = lb[1 * 32 + lane];
        v2f b2 = lb[2 * 32 + lane];
        v2f b3 = lb[3 * 32 + lane];
        v2f b4 = lb[4 * 32 + lane];
        x0 = __builtin_amdgcn_wmma_f32_16x16x4_f32(false, a, false, b0, (short)0, x0, false, false);
        y0 = __builtin_amdgcn_wmma_f32_16x16x4_f32(false, c, false, b0, (short)0, y0, false, false);
        x1 = __builtin_amdgcn_wmma_f32_16x16x4_f32(false, a, false, b1, (short)0, x1, false, false);
        y1 = __builtin_amdgcn_wmma_f32_16x16x4_f32(false, c, false, b1, (short)0, y1, false, false);
        x2 = __builtin_amdgcn_wmma_f32_16x16x4_f32(false, a, false, b2, (short)0, x2, false, false);
        y2 = __builtin_amdgcn_wmma_f32_16x16x4_f32(false, c, false, b2, (short)0, y2, false, false);
        x3 = __builtin_amdgcn_wmma_f32_16x16x4_f32(false, a, false, b3, (short)0, x3, false, false);
        y3 = __builtin_amdgcn_wmma_f32_16x16x4_f32(false, c, false, b3, (short)0, y3, false, false);
        x4 = __builtin_amdgcn_wmma_f32_16x16x4_f32(false, a, false, b4, (short)0, x4, false, false);
        y4 = __builtin_amdgcn_wmma_f32_16x16x4_f32(false, c, false, b4, (short)0, y4, false, false);
    }

    // Per-pixel inverse norm: lanes L and L^16 together cover all K of a row.
    ss0 += __shfl_xor(ss0, 16, 32);
    ss1 += __shfl_xor(ss1, 16, 32);
    float invn0 = 1.0f / fmaxf(sqrtf(ss0), 1e-12f);
    float invn1 = 1.0f / fmaxf(sqrtf(ss1), 1e-12f);
    int cls = gt[b * NPIX + n0 + lane];   // lane L -> class of pixel n0+L (32 pixels)

    v8f accX[5] = {x0, x1, x2, x3, x4};
    v8f accY[5] = {y0, y1, y2, y3, y4};
    #pragma unroll
    for (int r = 0; r < 8; ++r) {
        int M = r + (hi << 3);                 // pixel row held by this lane in vgpr r
        // tile 0 (pixels n0+M)
        float i0 = __shfl(invn0, M, 32);
        int   c0 = __shfl(cls,   M, 32);
        // tile 1 (pixels n0+16+M)
        float i1 = __shfl(invn1, M, 32);
        int   c1 = __shfl(cls,   16 + M, 32);
        #pragma unroll
        for (int t = 0; t < 5; ++t) {
            int J  = t * 16 + m;               // global proto column
            int q0 = J - c0 * NPROT;
            if (q0 >= 0 && q0 < NPROT)
                sims[(b * NPROT + q0) * NPIX + n0 + M] = accX[t][r] * i0;
            int q1 = J - c1 * NPROT;
            if (q1 >= 0 && q1 < NPROT)
                sims[(b * NPROT + q1) * NPIX + n0 + 16 + M] = accY[t][r] * i1;
        }
    }
}

// ---- pass: per-(b,c,p) max of sims over class pixels + class counts ----
__global__ void k_maxcnt(const float* __restrict__ sims, const int* __restrict__ gt,
                         unsigned* __restrict__ gmax, int* __restrict__ gcnt) {
    __shared__ unsigned mb[NPT];
    __shared__ int cb[NCLS];
    int tid = threadIdx.x;
    if (tid < NPT) mb[tid] = 0u;
    if (tid < NCLS) cb[tid] = 0;
    __syncthreads();
    int b = blockIdx.x >> 4, chunk = blockIdx.x & 15;
    int nb = chunk * 4096;
    for (int it = 0; it < 16; ++it) {
        int n = nb + it * 256 + tid;
        int c = gt[b * NPIX + n];
        atomicAdd(&cb[c], 1);
        #pragma unroll
        for (int q = 0; q < NPROT; ++q) {
            float v = sims[(b * NPROT + q) * NPIX + n];
            atomicMax(&mb[c * NPROT + q], mapf(v));
        }
    }
    __syncthreads();
    if (tid < NPT)  atomicMax(&gmax[b * NPT + tid], mb[tid]);
    if (tid < NCLS) atomicAdd(&gcnt[b * NCLS + tid], cb[tid]);
}

// ---- pass: Z1 = sum exp(sim - max1) over class pixels ----
__global__ void k_z1(const float* __restrict__ sims, const int* __restrict__ gt,
                     const unsigned* __restrict__ gmax, float* __restrict__ gZ1) {
    __shared__ float zb[NPT];
    int tid = threadIdx.x;
    if (tid < NPT) zb[tid] = 0.f;
    __syncthreads();
    int b = blockIdx.x >> 4, chunk = blockIdx.x & 15;
    int nb = chunk * 4096;
    for (int it = 0; it < 16; ++it) {
        int n = nb + it * 256 + tid;
        int c = gt[b * NPIX + n];
        #pragma unroll
        for (int q = 0; q < NPROT; ++q) {
            int idx = c * NPROT + q;
            float m1 = unmapf(gmax[b * NPT + idx]);
            float v  = sims[(b * NPROT + q) * NPIX + n];
            atomicAdd(&zb[idx], expf(v - m1));
        }
    }
    __syncthreads();
    if (tid < NPT) atomicAdd(&gZ1[b * NPT + tid], zb[tid]);
}

// ---- pass: Z2 = sum exp(s1 - 1/Z1), s1 = exp(sim - max1)/Z1 ----
__global__ void k_z2(const float* __restrict__ sims, const int* __restrict__ gt,
                     const unsigned* __restrict__ gmax, const float* __restrict__ gZ1,
                     float* __restrict__ gZ2) {
    __shared__ float zb[NPT];
    int tid = threadIdx.x;
    if (tid < NPT) zb[tid] = 0.f;
    __syncthreads();
    int b = blockIdx.x >> 4, chunk = blockIdx.x & 15;
    int nb = chunk * 4096;
    for (int it = 0; it < 16; ++it) {
        int n = nb + it * 256 + tid;
        int c = gt[b * NPIX + n];
        #pragma unroll
        for (int q = 0; q < NPROT; ++q) {
            int idx = c * NPROT + q;
            float m1 = unmapf(gmax[b * NPT + idx]);
            float r1 = 1.0f / gZ1[b * NPT + idx];
            float v  = sims[(b * NPROT + q) * NPIX + n];
            float s1 = expf(v - m1) * r1;
            atomicAdd(&zb[idx], expf(s1 - r1));
        }
    }
    __syncthreads();
    if (tid < NPT) atomicAdd(&gZ2[b * NPT + tid], zb[tid]);
}

// ---- pass: Gram G[b,c,i,j] = sum logV_i * V_j over class pixels ----
__global__ void k_gram(const float* __restrict__ sims, const int* __restrict__ gt,
                       const unsigned* __restrict__ gmax, const float* __restrict__ gZ1,
                       const float* __restrict__ gZ2, float* __restrict__ gG) {
    __shared__ float gb[NCLS * 25];
    int tid = threadIdx.x;
    for (int i = tid; i < NCLS * 25; i += blockDim.x) gb[i] = 0.f;
    __syncthreads();
    int b = blockIdx.x >> 5, chunk = blockIdx.x & 31;   // 128 blocks, 2048 pix each
    int nb = chunk * 2048;
    for (int it = 0; it < 8; ++it) {
        int n = nb + it * 256 + tid;
        int c = gt[b * NPIX + n];
        float V[NPROT], L[NPROT];
        #pragma unroll
        for (int q = 0; q < NPROT; ++q) {
            int idx = c * NPROT + q;
            float m1 = unmapf(gmax[b * NPT + idx]);
            float r1 = 1.0f / gZ1[b * NPT + idx];
            float r2 = 1.0f / gZ2[b * NPT + idx];
            float v  = sims[(b * NPROT + q) * NPIX + n];
            float s1 = expf(v - m1) * r1;
            float s2 = expf(s1 - r1) * r2;
            V[q] = s2 + EPSV;
            L[q] = logf(V[q]);
        }
        #pragma unroll
        for (int i = 0; i < NPROT; ++i)
            #pragma unroll
            for (int j = 0; j < NPROT; ++j)
                atomicAdd(&gb[c * 25 + i * 5 + j], L[i] * V[j]);
    }
    __syncthreads();
    for (int i = tid; i < NCLS * 25; i += blockDim.x)
        atomicAdd(&gG[b * NCLS * 25 + i], gb[i]);
}

// ---- finalize: J = M + M^T from G, simmat, class/batch averaging ----
__global__ void k_final(const float* __restrict__ gG, const int* __restrict__ gcnt,
                        float* __restrict__ out) {
    __shared__ float sumB[BATCH];
    __shared__ int   cntB[BATCH];
    int tid = threadIdx.x;
    if (tid < BATCH) { sumB[tid] = 0.f; cntB[tid] = 0; }
    __syncthreads();
    if (tid < BATCH * NCLS) {
        int b = tid >> 4, c = tid & 15;
        if (gcnt[tid] > 0) {
            const float* Gm = gG + b * NCLS * 25 + c * 25;
            float cs = 0.f;
            #pragma unroll
            for (int i = 0; i < NPROT; ++i)
                for (int j = i + 1; j < NPROT; ++j) {
                    float Jij = (Gm[j * 5 + j] - Gm[i * 5 + j]) +
                                (Gm[i * 5 + i] - Gm[j * 5 + i]);
                    cs += expf(-Jij);
                }
            atomicAdd(&sumB[b], cs * 0.1f);   // / num_pairs (10)
            atomicAdd(&cntB[b], 1);
        }
    }
    __syncthreads();
    if (tid == 0) {
        float loss = 0.f;
        for (int b = 0; b < BATCH; ++b)
            if (cntB[b] > 0) loss += sumB[b] / (float)cntB[b];
        out[0] = loss / (float)BATCH;
    }
}

extern "C" void kernel_launch(void* const* d_in, const int* in_sizes, int n_in,
                              void* d_out, int out_size, void* d_ws, size_t ws_size,
                              hipStream_t stream) {
    const float* feat   = (const float*)d_in[0];   // (B, D, H, W)
    const float* protos = (const float*)d_in[1];   // (C*P, D)
    const int*   gt     = (const int*)d_in[2];     // (B, H, W)
    float* out = (float*)d_out;

    // workspace carve-up
    float*    protoB = (float*)d_ws;                       // 20480 f
    float*    sims   = protoB + PROTOB_FLOATS;             // 1,310,720 f
    unsigned* gmax   = (unsigned*)(sims + BATCH * NPROT * NPIX);  // 320
    float*    gZ1    = (float*)(gmax + BATCH * NPT);       // 320
    float*    gZ2    = gZ1 + BATCH * NPT;                  // 320
    int*      gcnt   = (int*)(gZ2 + BATCH * NPT);          // 64
    float*    gG     = (float*)(gcnt + BATCH * NCLS);      // 1600

    k_init<<<7, 256, 0, stream>>>(gmax, gZ1, gZ2, gcnt, gG, out);
    k_protos<<<NPT, 256, 0, stream>>>(protos, protoB);
    // 262144 pixels / 32 per wave / 8 waves per block = 1024 blocks; 80KB dyn LDS
    k_sims<<<1024, 256, PROTOB_FLOATS * sizeof(float), stream>>>(feat, protoB, gt, sims);
    k_maxcnt<<<64, 256, 0, stream>>>(sims, gt, gmax, gcnt);
    k_z1<<<64, 256, 0, stream>>>(sims, gt, gmax, gZ1);
    k_z2<<<64, 256, 0, stream>>>(sims, gt, gmax, gZ1, gZ2);
    k_gram<<<128, 256, 0, stream>>>(sims, gt, gmax, gZ1, gZ2, gG);
    k_final<<<1, 256, 0, stream>>>(gG, gcnt, out);
}